// SimpleGraphConvolution_49443663512124
// MI455X (gfx1250) — compile-verified
//
#include <hip/hip_runtime.h>
#include <hip/hip_bf16.h>
#include <stdint.h>

#define D_FEAT 128

typedef __attribute__((ext_vector_type(2))) float v2f;
typedef __attribute__((ext_vector_type(8))) float v8f;

// ---------- deterministic per-element RNG (splitmix32-style) ----------
__device__ __forceinline__ uint32_t hash_u32(uint32_t x) {
    x ^= x >> 16; x *= 0x7feb352dU;
    x ^= x >> 15; x *= 0x846ca68bU;
    x ^= x >> 16;
    return x;
}

// ---------- hardware f32 atomic add (avoid CAS-loop fallback) ----------
__device__ __forceinline__ void atomic_add_f32(float* p, float v) {
    asm volatile("global_atomic_add_f32 %0, %1, off" : : "v"(p), "v"(v) : "memory");
}

// ---------- dropout: out[i] = keep ? in[i]*scale : 0 ----------
__global__ void dropout_kernel(const float* __restrict__ in, float* __restrict__ out,
                               int n, uint32_t seed, uint32_t thresh, float scale) {
    int i = blockIdx.x * blockDim.x + threadIdx.x;
    if (i >= n) return;
    uint32_t h = hash_u32((uint32_t)i * 0x9E3779B9u + seed);
    out[i] = (h < thresh) ? in[i] * scale : 0.0f;
}

__global__ void zero_kernel(float* __restrict__ p, int n) {
    int i = blockIdx.x * blockDim.x + threadIdx.x;
    if (i < n) p[i] = 0.0f;
}

// ---------- SpMM hop: one wave32 per edge; lane handles 4 of 128 feats ----------
__global__ void spmm_scatter_kernel(const float* __restrict__ xin,
                                    float* __restrict__ xout,
                                    const int* __restrict__ rows,
                                    const int* __restrict__ cols,
                                    const float* __restrict__ vals,
                                    int nedges) {
    int wave = (blockIdx.x * blockDim.x + threadIdx.x) >> 5;
    int lane = threadIdx.x & 31;
    if (wave >= nedges) return;
    float s = vals[wave];           // wave-uniform
    if (s == 0.0f) return;          // dropped edge: retire whole wave
    int c = cols[wave];
    int r = rows[wave];
    const float4 g = *(const float4*)(xin + (size_t)c * D_FEAT + lane * 4);
    float* dst = xout + (size_t)r * D_FEAT + lane * 4;
    atomic_add_f32(dst + 0, s * g.x);
    atomic_add_f32(dst + 1, s * g.y);
    atomic_add_f32(dst + 2, s * g.z);
    atomic_add_f32(dst + 3, s * g.w);
}

// ---------- out = relu(x @ w) via V_WMMA_F32_16X16X4_F32 ----------
// One wave per 16-row tile. A: 16x4 f32 (2 VGPRs/lane), B: w[k..k+3] broadcast
// across all 16 columns. Since B columns are identical, every column of the
// 16x16 f32 accumulator holds the dot product; extract per documented C/D
// layout (lane 0 -> M=0..7 in VGPRs 0..7, lane 16 -> M=8..15).
__global__ void gemv_relu_wmma_kernel(const float* __restrict__ x,
                                      const float* __restrict__ w,
                                      float* __restrict__ out, int ntiles) {
    __shared__ float ws[D_FEAT];
    int t = threadIdx.x;
    if (t < D_FEAT) ws[t] = w[t];
    __syncthreads();

    int wave = t >> 5;
    int lane = t & 31;
    int tile = blockIdx.x * (blockDim.x >> 5) + wave;  // wave-uniform guard
    if (tile >= ntiles) return;

    int half = lane >> 4;                 // 0: K slots {0,1}; 1: K slots {2,3}
    int m = (tile << 4) + (lane & 15);
    const float2* rp = (const float2*)(x + (size_t)m * D_FEAT);
    const float2* wp = (const float2*)ws;

    v8f c = {0.f, 0.f, 0.f, 0.f, 0.f, 0.f, 0.f, 0.f};
#pragma unroll 4
    for (int k2 = 0; k2 < D_FEAT / 2; k2 += 2) {  // k2 = k/2, k steps of 4
        float2 av = rp[k2 + half];
        float2 bv = wp[k2 + half];
        v2f a = {av.x, av.y};
        v2f b = {bv.x, bv.y};
        c = __builtin_amdgcn_wmma_f32_16x16x4_f32(false, a, false, b,
                                                  (short)0, c, false, false);
    }

    if ((lane & 15) == 0) {
        int mbase = (tile << 4) + half * 8;
#pragma unroll
        for (int r = 0; r < 8; ++r)
            out[mbase + r] = fmaxf(c[r], 0.0f);
    }
}

extern "C" void kernel_launch(void* const* d_in, const int* in_sizes, int n_in,
                              void* d_out, int out_size, void* d_ws, size_t ws_size,
                              hipStream_t stream) {
    const float* inputs = (const float*)d_in[0];
    const int*   rows   = (const int*)d_in[1];
    const int*   cols   = (const int*)d_in[2];
    const float* vals   = (const float*)d_in[3];
    const float* w      = (const float*)d_in[4];

    const int N = in_sizes[0] / D_FEAT;   // 50000
    const int E = in_sizes[1];            // 800000
    const int ND = N * D_FEAT;            // 6.4M floats

    float* xa = (float*)d_ws;             // 25.6 MB
    float* xb = xa + (size_t)ND;          // 25.6 MB
    float* vd = xb + (size_t)ND;          // 3.2 MB
    float* outp = (float*)d_out;

    const int BLK = 256;
    // keep-prob 0.8 -> thresh 0xCCCCCCCC, scale 1.25 ; keep-prob 0.5 -> 0x80000000, scale 2.0
    dropout_kernel<<<(ND + BLK - 1) / BLK, BLK, 0, stream>>>(inputs, xa, ND,
                                                             0xA341316Cu, 0xCCCCCCCCu, 1.25f);
    dropout_kernel<<<(E + BLK - 1) / BLK, BLK, 0, stream>>>(vals, vd, E,
                                                            0xC8013EA4u, 0x80000000u, 2.0f);
    zero_kernel<<<(ND + BLK - 1) / BLK, BLK, 0, stream>>>(xb, ND);

    const int waves_per_blk = BLK / 32;
    const int spmm_blocks = (E + waves_per_blk - 1) / waves_per_blk;

    // hop 1: xa -> xb
    spmm_scatter_kernel<<<spmm_blocks, BLK, 0, stream>>>(xa, xb, rows, cols, vd, E);
    // re-zero xa, hop 2: xb -> xa
    zero_kernel<<<(ND + BLK - 1) / BLK, BLK, 0, stream>>>(xa, ND);
    spmm_scatter_kernel<<<spmm_blocks, BLK, 0, stream>>>(xb, xa, rows, cols, vd, E);

    // relu(x @ w) with f32 WMMA
    const int ntiles = N / 16;  // 3125
    const int gblocks = (ntiles + waves_per_blk - 1) / waves_per_blk;
    gemv_relu_wmma_kernel<<<gblocks, BLK, 0, stream>>>(xa, w, outp, ntiles);
}